// SobelEdgeDetector_50981261804141
// MI455X (gfx1250) — compile-verified
//
#include <hip/hip_runtime.h>
#include <math.h>

// Sobel edge detector for x:(8,64,512,512) f32 -> out:(8,1,512,512) f32.
// Bandwidth-bound (~550MB traffic => ~24us floor at 23.3 TB/s on MI455X).
// Pass 1 uses V_WMMA_F32_16X16X4_F32 for the 64-channel mean reduction
// (full f32 precision; A is uniform 1/64 so the result is invariant to the
// exact B-fragment K<->lane mapping).

typedef float v2f __attribute__((ext_vector_type(2)));
typedef float v8f __attribute__((ext_vector_type(8)));

#define HH   512
#define WW   512
#define CHW  (512 * 512)          // pixels per (b, c) plane
#define NB   8                    // batches
#define NC   64                   // channels

// ---------------------------------------------------------------------------
// Pass 0: init per-batch min/max slots (min=+inf bits, max=0). mag > 0 always,
// so uint ordering == float ordering for the atomics in pass 2.
// ---------------------------------------------------------------------------
__global__ void sobel_init_minmax(unsigned int* __restrict__ minmax) {
    int i = threadIdx.x;               // 16 threads: [2b]=min, [2b+1]=max
    if (i < 2 * NB) minmax[i] = (i & 1) ? 0u : 0x7F800000u;
}

// ---------------------------------------------------------------------------
// Pass 1: channel mean via WMMA f32 16x16x4.
// One wave per 16-pixel group. B[k][n] = x[b][c+k][q0+n], A[m][k] = 1/64.
// Accumulate 16 WMMAs (k = 4 channels each) -> D rows all equal the mean.
// ---------------------------------------------------------------------------
__global__ __launch_bounds__(256) void sobel_mean_wmma(
        const float* __restrict__ x, float* __restrict__ mean) {
    const int lane  = threadIdx.x & 31;
    const int wave  = threadIdx.x >> 5;
    const int group = blockIdx.x * 8 + wave;     // 16-pixel group id
    const int b     = group >> 14;               // / (CHW/16 = 16384)
    const int q0    = (group & 16383) << 4;      // first pixel of group
    const int half  = lane >> 4;                 // B frag: lanes 16-31 hold K+2
    const int n     = lane & 15;                 // pixel within group

    const float* base = x + (size_t)b * NC * CHW + (size_t)q0 + n;

    v2f a;  a.x = 1.0f / 64.0f;  a.y = 1.0f / 64.0f;   // uniform A matrix
    v8f acc = {};

#pragma unroll
    for (int c = 0; c < NC; c += 4) {
        v2f bf;
        bf.x = base[(size_t)(c + 2 * half)     * CHW];
        bf.y = base[(size_t)(c + 2 * half + 1) * CHW];
        // D = A(16x4) * B(4x16) + C ; emits v_wmma_f32_16x16x4_f32
        acc = __builtin_amdgcn_wmma_f32_16x16x4_f32(
                /*neg_a=*/false, a, /*neg_b=*/false, bf,
                /*c_mod=*/(short)0, acc, /*reuse_a=*/false, /*reuse_b=*/false);
    }

    // C/D layout: VGPR0, lanes 0-15 -> M=0, N=lane : the 16 per-pixel means.
    if (lane < 16) mean[(size_t)b * CHW + q0 + lane] = acc[0];
}

// ---------------------------------------------------------------------------
// Pass 2: 3x3 Sobel (zero-padded SAME), magnitude, per-batch min/max.
// Sign flips from conv-vs-correlation cancel under the square.
// ---------------------------------------------------------------------------
__global__ __launch_bounds__(256) void sobel_mag_minmax(
        const float* __restrict__ mean, float* __restrict__ mag,
        unsigned int* __restrict__ minmax) {
    const int b = blockIdx.y;
    const int q = blockIdx.x * 256 + threadIdx.x;
    const int h = q >> 9;
    const int w = q & 511;
    const float* m = mean + (size_t)b * CHW;

    float v[3][3];
#pragma unroll
    for (int i = -1; i <= 1; ++i)
#pragma unroll
        for (int j = -1; j <= 1; ++j) {
            int hh = h + i, ww = w + j;
            v[i + 1][j + 1] = (hh >= 0 && hh < HH && ww >= 0 && ww < WW)
                                  ? m[hh * WW + ww] : 0.0f;
        }

    float gx = (v[0][2] - v[0][0]) + 2.0f * (v[1][2] - v[1][0]) + (v[2][2] - v[2][0]);
    float gy = (v[2][0] - v[0][0]) + 2.0f * (v[2][1] - v[0][1]) + (v[2][2] - v[0][2]);
    float mg = sqrtf(fmaf(gx, gx, fmaf(gy, gy, 1e-8f)));

    mag[(size_t)b * CHW + q] = mg;

    // wave32 shuffle reduction, then LDS across the 8 waves, then atomics.
    float mn = mg, mx = mg;
#pragma unroll
    for (int off = 16; off > 0; off >>= 1) {
        mn = fminf(mn, __shfl_xor(mn, off, 32));
        mx = fmaxf(mx, __shfl_xor(mx, off, 32));
    }
    __shared__ float smn[8], smx[8];
    const int lane = threadIdx.x & 31, wv = threadIdx.x >> 5;
    if (lane == 0) { smn[wv] = mn; smx[wv] = mx; }
    __syncthreads();
    if (threadIdx.x == 0) {
        float bmn = smn[0], bmx = smx[0];
#pragma unroll
        for (int i = 1; i < 8; ++i) {
            bmn = fminf(bmn, smn[i]);
            bmx = fmaxf(bmx, smx[i]);
        }
        // mg > 0 always -> uint bit ordering matches float ordering.
        atomicMin(&minmax[2 * b + 0], __float_as_uint(bmn));
        atomicMax(&minmax[2 * b + 1], __float_as_uint(bmx));
    }
}

// ---------------------------------------------------------------------------
// Pass 3: normalize in place on d_out.
// ---------------------------------------------------------------------------
__global__ __launch_bounds__(256) void sobel_norm(
        float* __restrict__ out, const unsigned int* __restrict__ minmax) {
    const int b = blockIdx.y;
    const size_t idx = (size_t)b * CHW + blockIdx.x * 256 + threadIdx.x;
    const float mn  = __uint_as_float(minmax[2 * b + 0]);
    const float mx  = __uint_as_float(minmax[2 * b + 1]);
    const float inv = 1.0f / (mx - mn + 1e-8f);
    out[idx] = (out[idx] - mn) * inv;
}

// ---------------------------------------------------------------------------
extern "C" void kernel_launch(void* const* d_in, const int* in_sizes, int n_in,
                              void* d_out, int out_size, void* d_ws, size_t ws_size,
                              hipStream_t stream) {
    (void)in_sizes; (void)n_in; (void)out_size; (void)ws_size;

    const float* x   = (const float*)d_in[0];
    float*       out = (float*)d_out;                         // mag, then normed
    float*       mean = (float*)d_ws;                          // 8 MiB
    unsigned int* minmax =
        (unsigned int*)((char*)d_ws + (size_t)NB * CHW * sizeof(float));

    sobel_init_minmax<<<1, 16, 0, stream>>>(minmax);

    // 8*CHW/16 = 131072 pixel-groups, 8 waves (groups) per 256-thread block.
    sobel_mean_wmma<<<131072 / 8, 256, 0, stream>>>(x, mean);

    dim3 g2(CHW / 256, NB);
    sobel_mag_minmax<<<g2, 256, 0, stream>>>(mean, out, minmax);
    sobel_norm<<<g2, 256, 0, stream>>>(out, minmax);
}